// UNet_5557687681339
// MI455X (gfx1250) — compile-verified
//
#include <hip/hip_runtime.h>
#include <math.h>

// ---------------- problem constants (match reference) ----------------
#define BN_  8
#define P0_  4096
#define P1_  2048
#define P2_  1024
#define D_   64
#define KNN_ 16
#define NCLS_ 13

typedef __attribute__((ext_vector_type(2))) float v2f;
typedef __attribute__((ext_vector_type(8))) float v8f;

#if defined(__gfx1250__) && __has_builtin(__builtin_amdgcn_tensor_load_to_lds)
#define HAS_TDM 1
#else
#define HAS_TDM 0
#endif

#if HAS_TDM
typedef __attribute__((ext_vector_type(4))) unsigned tdm_v4u;
typedef __attribute__((ext_vector_type(4))) int      tdm_v4i;
typedef __attribute__((ext_vector_type(8))) int      tdm_v8i;

// Issue one TDM 2D tile load: global (dim0 x dim1, stride0) -> LDS tile
// (tile0 x tile1), optional LDS padding (pad_interval code, pad_amount code).
__device__ __forceinline__ void tdm_load_2d(unsigned lds_addr, const void* gptr,
                                            unsigned dim0, unsigned dim1,
                                            unsigned tile0, unsigned tile1,
                                            unsigned stride0, unsigned stride1,
                                            unsigned pad_en, unsigned pad_int,
                                            unsigned pad_amt) {
  unsigned long long ga = (unsigned long long)gptr;
  tdm_v4u g0;
  g0.x = 1u;                                              // count=1 (valid D#)
  g0.y = lds_addr;                                        // LDS byte address
  g0.z = (unsigned)(ga & 0xFFFFFFFFu);                    // global_addr[31:0]
  g0.w = ((unsigned)(ga >> 32) & 0x01FFFFFFu) | (2u << 30); // addr[56:32]|type=2
  tdm_v8i g1;
  g1[0] = (int)((2u << 16) | (pad_en << 20) | (pad_int << 22) | (pad_amt << 25));
  g1[1] = (int)((dim0 & 0xFFFFu) << 16);                  // tensor_dim0 lo
  g1[2] = (int)((dim0 >> 16) | ((dim1 & 0xFFFFu) << 16)); // dim0 hi | dim1 lo
  g1[3] = (int)((dim1 >> 16) | (tile0 << 16));            // dim1 hi | tile_dim0
  g1[4] = (int)(tile1 & 0xFFFFu);                         // tile_dim1 (tile2=0)
  g1[5] = (int)stride0;                                   // dim0_stride lo32
  g1[6] = (int)((stride1 & 0xFFFFu) << 16);               // dim0 hi=0 | dim1 lo
  g1[7] = (int)(stride1 >> 16);                           // dim1_stride hi
  tdm_v4i z4 = {0, 0, 0, 0};
#if __clang_major__ >= 23
  tdm_v8i z8 = {0, 0, 0, 0, 0, 0, 0, 0};
  __builtin_amdgcn_tensor_load_to_lds(g0, g1, z4, z4, z8, 0);
#else
  __builtin_amdgcn_tensor_load_to_lds(g0, g1, z4, z4, 0);
#endif
}
#endif

// =====================================================================
// GEMM: Y[rows,64] = X[rows,64] @ W[64,64]  (+ optional bias & ELU)
// Block = 8 waves = 128 rows. W (16KB) and the 128x64 X tile (LDS stride
// 65 to avoid bank aliasing) staged via the Tensor Data Mover; each wave
// owns one 16-col tile with register-resident B fragments and sweeps 4
// row tiles of V_WMMA_F32_16X16X4_F32 (64 wmma / wave).
// mode 0: plain store; mode 1: y = elu(y + bias[col])
// =====================================================================
__global__ void gemm_rows64x64(const float* __restrict__ X,
                               const float* __restrict__ W,
                               const float* __restrict__ bias,
                               float* __restrict__ Y, int rows, int mode) {
  __shared__ float Wl[64 * 64];
  __shared__ float Xl[128 * 65];
  int tid = threadIdx.x;
  int blockRow = blockIdx.x * 128;
#if HAS_TDM
  if (tid < 32) {   // one wave issues both DMAs (TDM ignores EXEC)
    tdm_load_2d((unsigned)(unsigned long long)(void*)Wl, W,
                64, 64, 64, 64, 64, 64, 0, 0, 0);
    tdm_load_2d((unsigned)(unsigned long long)(void*)Xl, X + blockRow * 64,
                64, (unsigned)(rows - blockRow), 64, 128, 64, 64,
                1, 5, 0);                 // pad: +1 DWORD every 64 DWORDs
    __builtin_amdgcn_s_wait_tensorcnt(0);
  }
#else
  for (int i = tid; i < 4096; i += 256) Wl[i] = W[i];
  for (int i = tid; i < 8192; i += 256)
    Xl[(i >> 6) * 65 + (i & 63)] = X[blockRow * 64 + i];
#endif
  __syncthreads();

  int wave = tid >> 5, lane = tid & 31;
  int half = lane >> 4, lm = lane & 15;
  int ct = wave & 3;                    // column tile owned by this wave
  int rw = wave >> 2;                   // row-tile sub-index (0..1)

  // B fragments (4x16 K-major slabs of W), register resident for the kernel
  v2f bfrag[16];
#pragma unroll
  for (int s = 0; s < 16; ++s) {
    int k0 = 4 * s + 2 * half;
    bfrag[s].x = Wl[k0 * 64 + 16 * ct + lm];
    bfrag[s].y = Wl[(k0 + 1) * 64 + 16 * ct + lm];
  }

#pragma unroll
  for (int it = 0; it < 4; ++it) {
    int rl = it * 32 + rw * 16 + lm;    // this lane's A row inside LDS tile
    const float* ax = &Xl[rl * 65];
    v8f acc = {};
#pragma unroll
    for (int s = 0; s < 16; ++s) {
      int k0 = 4 * s + 2 * half;
      v2f a; a.x = ax[k0]; a.y = ax[k0 + 1];
      acc = __builtin_amdgcn_wmma_f32_16x16x4_f32(false, a, false, bfrag[s],
                                                  (short)0, acc, false, false);
    }
    int m0 = blockRow + it * 32 + rw * 16;
#pragma unroll
    for (int j = 0; j < 8; ++j) {
      int m = m0 + j + 8 * half;        // C/D: VGPR j -> M = j + 8*half
      float v = acc[j];
      if (mode == 1) {
        v += bias[16 * ct + lm];
        v = v > 0.f ? v : (__expf(v) - 1.f);
      }
      Y[m * 64 + 16 * ct + lm] = v;
    }
  }
}

// ================= embedding: feat[N,6] @ W_emb[6,64] =================
__global__ void embed6to64(const float* __restrict__ feat,
                           const float* __restrict__ W,
                           float* __restrict__ Y) {
  int t = blockIdx.x * 256 + threadIdx.x;   // rows*64 threads
  int row = t >> 6, ch = t & 63;
  const float* f = feat + row * 6;
  float acc = 0.f;
#pragma unroll
  for (int k = 0; k < 6; ++k) acc += f[k] * W[k * 64 + ch];
  Y[t] = acc;
}

// ============== BatchNorm (training stats) : two-pass =================
__global__ void bn_stats_partial(const float* __restrict__ Z,
                                 float* __restrict__ partial, int rows) {
  __shared__ float ssum[256], ssq[256];
  int ch = threadIdx.x & 63, rg = threadIdx.x >> 6;
  float s = 0.f, q = 0.f;
  for (int r = blockIdx.x * 4 + rg; r < rows; r += gridDim.x * 4) {
    float v = Z[r * 64 + ch];
    s += v; q += v * v;
  }
  ssum[threadIdx.x] = s; ssq[threadIdx.x] = q;
  __syncthreads();
  if (rg == 0) {
    s = ssum[ch] + ssum[64 + ch] + ssum[128 + ch] + ssum[192 + ch];
    q = ssq[ch] + ssq[64 + ch] + ssq[128 + ch] + ssq[192 + ch];
    partial[blockIdx.x * 128 + ch] = s;
    partial[blockIdx.x * 128 + 64 + ch] = q;
  }
}

__global__ void bn_stats_final(const float* __restrict__ partial,
                               float* __restrict__ stats, int nblocks, float inv_n) {
  int ch = threadIdx.x;  // 64 threads
  float s = 0.f, q = 0.f;
  for (int b = 0; b < nblocks; ++b) { s += partial[b * 128 + ch]; q += partial[b * 128 + 64 + ch]; }
  float mean = s * inv_n;
  float var = q * inv_n - mean * mean;
  stats[ch] = mean;
  stats[64 + ch] = rsqrtf(var + 1e-5f);
}

// mode 0: H = bn(Z); mode 1: H = relu(bn(Z)); mode 2: H = elu(bn(Z)) + H
__global__ void bn_apply(const float* __restrict__ Z, const float* __restrict__ stats,
                         const float* __restrict__ g, const float* __restrict__ b,
                         float* __restrict__ H, int mode) {
  int t = blockIdx.x * 256 + threadIdx.x;
  int ch = t & 63;
  float y = (Z[t] - stats[ch]) * stats[64 + ch] * g[ch] + b[ch];
  if (mode == 1) y = fmaxf(y, 0.f);
  else if (mode == 2) { y = (y > 0.f) ? y : (__expf(y) - 1.f); y += H[t]; }
  H[t] = y;
}

// ====================== KNN (per instance, K=16) ======================
__global__ void knn_kernel(const float* __restrict__ coords,
                           int* __restrict__ nbr, int Pp) {
  int tilesPer = Pp >> 8;
  int binst = blockIdx.x / tilesPer;
  int qtile = blockIdx.x % tilesPer;
  const float* cb = coords + binst * Pp * 3;
  int q = qtile * 256 + threadIdx.x;
  float qx = cb[q * 3], qy = cb[q * 3 + 1], qz = cb[q * 3 + 2];
  float bd[16]; int bi[16];
#pragma unroll
  for (int i = 0; i < 16; ++i) { bd[i] = 3.0e38f; bi[i] = 0; }
  __shared__ float tx[256], ty[256], tz[256];
  for (int tile = 0; tile < tilesPer; ++tile) {
    int j = tile * 256 + threadIdx.x;
    tx[threadIdx.x] = cb[j * 3];
    ty[threadIdx.x] = cb[j * 3 + 1];
    tz[threadIdx.x] = cb[j * 3 + 2];
    __syncthreads();
    if (tile + 1 < tilesPer)                      // gfx1250 global_prefetch
      __builtin_prefetch(cb + (tile + 1) * 768 + threadIdx.x * 3, 0, 0);
    int base = tile * 256;
#pragma unroll 4
    for (int u = 0; u < 256; ++u) {
      float dx = qx - tx[u], dy = qy - ty[u], dz = qz - tz[u];
      float d = dx * dx + dy * dy + dz * dz;
      int gj = base + u;
      if (d < bd[15] && gj != q) {                // self excluded (eye*1e9)
        bd[15] = d; bi[15] = gj;
#pragma unroll
        for (int s2 = 15; s2 > 0; --s2) {
          if (bd[s2] < bd[s2 - 1]) {
            float td = bd[s2]; bd[s2] = bd[s2 - 1]; bd[s2 - 1] = td;
            int ti = bi[s2]; bi[s2] = bi[s2 - 1]; bi[s2 - 1] = ti;
          }
        }
      }
    }
    __syncthreads();
  }
  int* out = nbr + (binst * Pp + q) * 16;
#pragma unroll
  for (int i = 0; i < 16; ++i) out[i] = bi[i];
}

// ======== downsample top-k: score + full bitonic sort in LDS ==========
__global__ void topk_kernel(const float* __restrict__ H,
                            const float* __restrict__ wscore,
                            int* __restrict__ idx_out, int Pp, int M) {
  __shared__ float sc[4096];
  __shared__ int   id[4096];
  int b = blockIdx.x;
  const float* hb = H + b * Pp * 64;
  for (int p = threadIdx.x; p < Pp; p += blockDim.x) {
    const float* hr = hb + p * 64;
    float s = 0.f;
#pragma unroll
    for (int k = 0; k < 64; ++k) s += hr[k] * wscore[k];
    sc[p] = s; id[p] = p;
  }
  __syncthreads();
  for (int k = 2; k <= Pp; k <<= 1)
    for (int j = k >> 1; j > 0; j >>= 1) {
      for (int i = threadIdx.x; i < Pp; i += blockDim.x) {
        int ixj = i ^ j;
        if (ixj > i) {
          bool dirDesc = ((i & k) == 0);
          float si = sc[i], sj = sc[ixj]; int ii = id[i], ij = id[ixj];
          bool iGt = (si > sj) || (si == sj && ii < ij);  // jax tie: lower idx
          if (dirDesc ? !iGt : iGt) { sc[i] = sj; sc[ixj] = si; id[i] = ij; id[ixj] = ii; }
        }
      }
      __syncthreads();
    }
  for (int m = threadIdx.x; m < M; m += blockDim.x) idx_out[b * M + m] = id[m];
}

// ====== neighbor gather-max:  Z = (mode? Z : 0-base) + max_k T[nbr] ===
__global__ void gathermax_kernel(const float* __restrict__ T,
                                 const int* __restrict__ nbr,
                                 float* __restrict__ Z, int Pp, int mode) {
  __shared__ int nb[4][16];
  int tid = threadIdx.x;
  int row0 = blockIdx.x * 4;
  if (tid < 64) nb[tid >> 4][tid & 15] = nbr[(row0 + (tid >> 4)) * 16 + (tid & 15)];
  __syncthreads();
  int ch = tid & 63, lp = tid >> 6;
  int row = row0 + lp;
  int base = (row / Pp) * Pp;                 // per-instance offset
  float m = -3.0e38f;
#pragma unroll
  for (int k = 0; k < 16; ++k) m = fmaxf(m, T[(base + nb[lp][k]) * 64 + ch]);
  int o = row * 64 + ch;
  Z[o] = mode ? (Z[o] + m) : m;
}

// ====================== misc data movement ============================
__global__ void zero_kernel(float* __restrict__ A) {
  A[blockIdx.x * 256 + threadIdx.x] = 0.f;
}
__global__ void add_kernel(float* __restrict__ A, const float* __restrict__ B) {
  int t = blockIdx.x * 256 + threadIdx.x; A[t] += B[t];
}
__global__ void gather_rows64(const float* __restrict__ In, const int* __restrict__ idx,
                              float* __restrict__ Out, int Pin, int M) {
  int t = blockIdx.x * 256 + threadIdx.x;   // B*M*64
  int r = t >> 6, ch = t & 63;
  int b = r / M;
  Out[t] = In[(b * Pin + idx[r]) * 64 + ch];
}
__global__ void gather_rows3(const float* __restrict__ In, const int* __restrict__ idx,
                             float* __restrict__ Out, int Pin, int M) {
  int t = blockIdx.x * 256 + threadIdx.x;   // B*M*3
  int r = t / 3, c = t - r * 3;
  int b = r / M;
  Out[t] = In[(b * Pin + idx[r]) * 3 + c];
}
__global__ void scatter_rows64(const float* __restrict__ In, const int* __restrict__ idx,
                               float* __restrict__ Out, int Pout, int M) {
  int t = blockIdx.x * 256 + threadIdx.x;   // B*M*64
  int r = t >> 6, ch = t & 63;
  int b = r / M;
  Out[(b * Pout + idx[r]) * 64 + ch] = In[t];
}

// =============== classifier: X[rows,64] @ Wc[64,13] ===================
__global__ void clust_kernel(const float* __restrict__ X, const float* __restrict__ Wc,
                             float* __restrict__ Out, int rows) {
  __shared__ float Wl[64 * 13];
  for (int i = threadIdx.x; i < 832; i += 256) Wl[i] = Wc[i];
  __syncthreads();
  int t = blockIdx.x * 256 + threadIdx.x;
  int row = t >> 4, c = t & 15;
  if (row < rows && c < 13) {
    const float* xr = X + row * 64;
    float s = 0.f;
#pragma unroll
    for (int k = 0; k < 64; ++k) s += xr[k] * Wl[k * 13 + c];
    Out[row * 13 + c] = s;
  }
}

// =====================================================================
extern "C" void kernel_launch(void* const* d_in, const int* in_sizes, int n_in,
                              void* d_out, int out_size, void* d_ws, size_t ws_size,
                              hipStream_t stream) {
  (void)in_sizes; (void)n_in; (void)out_size; (void)ws_size;
  const int rows0 = BN_ * P0_, rows1 = BN_ * P1_, rows2 = BN_ * P2_;

  // -------- inputs (setup_inputs order) --------
  const float* coord  = (const float*)d_in[0];
  const float* feat   = (const float*)d_in[1];
  const float* W_emb  = (const float*)d_in[2];
  const float* g_emb  = (const float*)d_in[3];
  const float* b_emb  = (const float*)d_in[4];
  const float* Ws0 = (const float*)d_in[5];  const float* Wm0 = (const float*)d_in[6];
  const float* g0  = (const float*)d_in[7];  const float* b0  = (const float*)d_in[8];
  const float* Ws1 = (const float*)d_in[9];  const float* Wm1 = (const float*)d_in[10];
  const float* g1  = (const float*)d_in[11]; const float* b1  = (const float*)d_in[12];
  const float* Ws2 = (const float*)d_in[13]; const float* Wm2 = (const float*)d_in[14];
  const float* g2  = (const float*)d_in[15]; const float* b2  = (const float*)d_in[16];
  const float* wsc0 = (const float*)d_in[17]; const float* wsc1 = (const float*)d_in[18];
  const float* Wsu0 = (const float*)d_in[19]; const float* Wmu0 = (const float*)d_in[20];
  const float* gu0  = (const float*)d_in[21]; const float* bu0  = (const float*)d_in[22];
  const float* Wsu1 = (const float*)d_in[23]; const float* Wmu1 = (const float*)d_in[24];
  const float* gu1  = (const float*)d_in[25]; const float* bu1  = (const float*)d_in[26];
  const float* W_post = (const float*)d_in[27]; const float* b_post = (const float*)d_in[28];
  const float* gamma_f = (const float*)d_in[29]; const float* beta_f = (const float*)d_in[30];
  const float* W_clust = (const float*)d_in[31];
  float* out = (float*)d_out;

  // -------- workspace carve-up (~50 MB) --------
  char* wp = (char*)d_ws;
  auto carve = [&](size_t bytes) { void* p = (void*)wp; wp += (bytes + 255) & ~(size_t)255; return p; };
  float* h0 = (float*)carve((size_t)rows0 * 64 * 4);
  float* h1 = (float*)carve((size_t)rows1 * 64 * 4);
  float* h2 = (float*)carve((size_t)rows2 * 64 * 4);
  float* hc = (float*)carve((size_t)rows0 * 64 * 4);
  float* z  = (float*)carve((size_t)rows0 * 64 * 4);
  float* tt = (float*)carve((size_t)rows0 * 64 * 4);
  float* up = (float*)carve((size_t)rows0 * 64 * 4);
  float* c1 = (float*)carve((size_t)rows1 * 3 * 4);
  float* c2 = (float*)carve((size_t)rows2 * 3 * 4);
  int* nbr0 = (int*)carve((size_t)rows0 * 16 * 4);
  int* nbr1 = (int*)carve((size_t)rows1 * 16 * 4);
  int* nbr2 = (int*)carve((size_t)rows2 * 16 * 4);
  int* idx0 = (int*)carve((size_t)rows1 * 4);
  int* idx1 = (int*)carve((size_t)rows2 * 4);
  float* partial = (float*)carve(64 * 128 * 4);
  float* stats   = (float*)carve(128 * 4);

  auto run_gemm = [&](const float* X, const float* W, const float* bias, float* Y,
                      int rows, int mode) {
    gemm_rows64x64<<<rows / 128, 256, 0, stream>>>(X, W, bias, Y, rows, mode);
  };
  auto run_bn = [&](const float* Z, const float* g, const float* b, float* H,
                    int rows, int mode) {
    bn_stats_partial<<<64, 256, 0, stream>>>(Z, partial, rows);
    bn_stats_final<<<1, 64, 0, stream>>>(partial, stats, 64, 1.0f / (float)rows);
    bn_apply<<<rows * 64 / 256, 256, 0, stream>>>(Z, stats, g, b, H, mode);
  };
  // h = elu(bn(h@Ws + max_k (h@Wm)[nbr])) + h, repeated nl times
  auto run_mp = [&](float* h, const int* nbr, const float* Ws, const float* Wm,
                    const float* g, const float* b, int nl, int rows, int Pp) {
    for (int l = 0; l < nl; ++l) {
      run_gemm(h, Ws + l * 4096, nullptr, z, rows, 0);   // self transform -> z
      run_gemm(h, Wm + l * 4096, nullptr, tt, rows, 0);  // message transform -> tt
      gathermax_kernel<<<rows / 4, 256, 0, stream>>>(tt, nbr, z, Pp, 1);  // z += max_k
      run_bn(z, g + l * 64, b + l * 64, h, rows, 2);     // h = elu(bn(z)) + h
    }
  };

  // ---------------- embedding: Linear -> BN -> ReLU ----------------
  embed6to64<<<rows0 * 64 / 256, 256, 0, stream>>>(feat, W_emb, z);
  run_bn(z, g_emb, b_emb, h0, rows0, 1);

  // ---------------- down path ----------------
  knn_kernel<<<BN_ * (P0_ / 256), 256, 0, stream>>>(coord, nbr0, P0_);
  run_mp(h0, nbr0, Ws0, Wm0, g0, b0, 2, rows0, P0_);
  topk_kernel<<<BN_, 512, 0, stream>>>(h0, wsc0, idx0, P0_, P1_);
  gather_rows3<<<rows1 * 3 / 256, 256, 0, stream>>>(coord, idx0, c1, P0_, P1_);
  gather_rows64<<<rows1 * 64 / 256, 256, 0, stream>>>(h0, idx0, h1, P0_, P1_);
  knn_kernel<<<BN_ * (P1_ / 256), 256, 0, stream>>>(c1, nbr1, P1_);
  run_mp(h1, nbr1, Ws1, Wm1, g1, b1, 4, rows1, P1_);
  topk_kernel<<<BN_, 512, 0, stream>>>(h1, wsc1, idx1, P1_, P2_);
  gather_rows3<<<rows2 * 3 / 256, 256, 0, stream>>>(c1, idx1, c2, P1_, P2_);
  gather_rows64<<<rows2 * 64 / 256, 256, 0, stream>>>(h1, idx1, h2, P1_, P2_);
  knn_kernel<<<BN_ * (P2_ / 256), 256, 0, stream>>>(c2, nbr2, P2_);
  run_mp(h2, nbr2, Ws2, Wm2, g2, b2, 4, rows2, P2_);     // bottleneck

  // ---------------- up path: scatter + neighbor max + MP + skip ----------------
  zero_kernel<<<rows1 * 64 / 256, 256, 0, stream>>>(up);
  scatter_rows64<<<rows2 * 64 / 256, 256, 0, stream>>>(h2, idx1, up, P1_, P2_);
  gathermax_kernel<<<rows1 / 4, 256, 0, stream>>>(up, nbr1, hc, P1_, 0);
  run_mp(hc, nbr1, Wsu0, Wmu0, gu0, bu0, 4, rows1, P1_);
  add_kernel<<<rows1 * 64 / 256, 256, 0, stream>>>(hc, h1);

  zero_kernel<<<rows0 * 64 / 256, 256, 0, stream>>>(up);
  scatter_rows64<<<rows1 * 64 / 256, 256, 0, stream>>>(hc, idx0, up, P0_, P1_);
  gathermax_kernel<<<rows0 / 4, 256, 0, stream>>>(up, nbr0, hc, P0_, 0);
  run_mp(hc, nbr0, Wsu1, Wmu1, gu1, bu1, 2, rows0, P0_);
  add_kernel<<<rows0 * 64 / 256, 256, 0, stream>>>(hc, h0);

  // ---------------- head: 3x(Linear+ELU) -> BN -> classifier ----------------
  run_gemm(hc, W_post + 0 * 4096, b_post + 0 * 64, z,  rows0, 1);
  run_gemm(z,  W_post + 1 * 4096, b_post + 1 * 64, tt, rows0, 1);
  run_gemm(tt, W_post + 2 * 4096, b_post + 2 * 64, z,  rows0, 1);
  run_bn(z, gamma_f, beta_f, z, rows0, 0);
  clust_kernel<<<rows0 * 16 / 256, 256, 0, stream>>>(z, W_clust, out, rows0);
}